// AttentionGNN_13752485282260
// MI455X (gfx1250) — compile-verified
//
#include <hip/hip_runtime.h>
#include <cstdint>

// ---------------------------------------------------------------------------
// A3TGCN on MI455X (gfx1250, wave32).
//
// Strategy:
//   1) gcn(x,W) = A_hat (x W) = (A_hat x) W  -> hoist ONE SpMM over all
//      T*F_IN = 192 features (output layout [T][N][16] for clean GEMMs).
//      Scatter-add with global_atomic_add_f32; src rows + dst matrix are
//      L2-resident (76.8MB each vs 192MB L2).
//   2) Per time step: fused TGCN cell kernel. Each wave owns a 16-node tile
//      and runs all GEMMs with V_WMMA_F32_16X16X4_F32 (f32 path keeps the
//      12-step sigmoid/tanh recurrence numerically faithful).
//      Linear([G,h]) = G @ W[0:32] + h @ W[32:64]  (no concat materialized).
//      D-layout -> A-layout relayout goes through a private per-wave LDS tile.
//   3) Tiny head kernel: relu -> [32x2] -> softmax2.
// ---------------------------------------------------------------------------

#define F_IN 16
#define HID  32
#define LDSS 34   // padded LDS row stride (floats) to dodge bank conflicts

typedef __attribute__((ext_vector_type(2))) float v2f;
typedef __attribute__((ext_vector_type(8))) float v8f;

__device__ __forceinline__ v8f wmma4(v2f a, v2f b, v8f c) {
  // D = A(16x4,f32) * B(4x16,f32) + C(16x16,f32)
  return __builtin_amdgcn_wmma_f32_16x16x4_f32(false, a, false, b, (short)0, c,
                                               false, false);
}
__device__ __forceinline__ v8f splat8(float v) {
  v8f r = {v, v, v, v, v, v, v, v};
  return r;
}
// A chunk (16 rows x 4 K) from a row-major [16 x ld] tile.
// Lanes 0-15: M=lane, K = 4kb+{0,1}; lanes 16-31: M=lane-16, K = 4kb+{2,3}.
__device__ __forceinline__ v2f a_chunk(const float* M, int ld, int row, int hi,
                                       int kb) {
  const float* p = M + row * ld + kb * 4 + hi * 2;
  v2f a = {p[0], p[1]};
  return a;
}
// B chunk (4 K x 16 cols) from a row-major [K x 32] weight matrix.
__device__ __forceinline__ v2f b_chunk(const float* W, int kb, int hi, int col) {
  const int k = kb * 4 + hi * 2;
  v2f b = {W[k * HID + col], W[(k + 1) * HID + col]};
  return b;
}
__device__ __forceinline__ void ds_fence() {
  __asm__ volatile("s_wait_dscnt 0" ::: "memory");
}

// ------------------------------- utility kernels ---------------------------

__global__ void fill_f32(float* __restrict__ p, float v, long long n) {
  long long i = (long long)blockIdx.x * blockDim.x + threadIdx.x;
  long long stride = (long long)gridDim.x * blockDim.x;
  for (; i < n; i += stride) p[i] = v;
}

__global__ void deg_accum(const int* __restrict__ dst, float* __restrict__ deg,
                          long long E) {
  long long i = (long long)blockIdx.x * blockDim.x + threadIdx.x;
  long long stride = (long long)gridDim.x * blockDim.x;
  for (; i < E; i += stride) atomicAdd(&deg[dst[i]], 1.0f);
}

__global__ void rsqrt_k(const float* __restrict__ deg, float* __restrict__ dinv,
                        int n) {
  int i = blockIdx.x * blockDim.x + threadIdx.x;
  if (i < n) dinv[i] = rsqrtf(deg[i]);
}

__global__ void attn_softmax_k(const float* __restrict__ att,
                               float* __restrict__ probs, int T) {
  if (blockIdx.x == 0 && threadIdx.x == 0) {
    float m = -3.0e38f;
    for (int i = 0; i < T; ++i) m = fmaxf(m, att[i]);
    float s = 0.0f;
    for (int i = 0; i < T; ++i) s += __expf(att[i] - m);
    float inv = 1.0f / s;
    for (int i = 0; i < T; ++i) probs[i] = __expf(att[i] - m) * inv;
  }
}

// ----------------------- SpMM: AX[t][n][f] = (A_hat X) ---------------------
// Edge e in [0,E): src/dst from edge list, norm = dinv[s]*dinv[d].
// Edge e in [E,E+N): self loop, norm = dinv[n]^2.
// Each work item handles one float4 chunk (one feature f, 4 consecutive t).
__global__ void spmm_scatter(const int* __restrict__ src,
                             const int* __restrict__ dst,
                             const float* __restrict__ x,
                             const float* __restrict__ dinv,
                             float* __restrict__ AX, long long E, int n, int T) {
  const int cpt = T / 4;           // float4 chunks per feature (T=12 -> 3)
  const int cpe = F_IN * cpt;      // chunks per edge (48)
  const long long total = (E + (long long)n) * (long long)cpe;
  const long long stride = (long long)gridDim.x * blockDim.x;
  const size_t ts = (size_t)n * F_IN;  // per-time-step plane size
  for (long long w = (long long)blockIdx.x * blockDim.x + threadIdx.x;
       w < total; w += stride) {
    long long e = w / cpe;
    int c = (int)(w - e * cpe);
    int s, d;
    float nm;
    if (e < E) {
      s = src[e];
      d = dst[e];
      nm = dinv[s] * dinv[d];
    } else {
      s = d = (int)(e - E);
      float di = dinv[s];
      nm = di * di;
    }
    int f = c / cpt;
    int q = c - f * cpt;
    const float4 v =
        *(const float4*)(x + (size_t)s * (F_IN * T) + (size_t)f * T + q * 4);
    const int t0 = q * 4;
    const size_t base = (size_t)d * F_IN + f;
    atomicAdd(AX + (size_t)(t0 + 0) * ts + base, nm * v.x);
    atomicAdd(AX + (size_t)(t0 + 1) * ts + base, nm * v.y);
    atomicAdd(AX + (size_t)(t0 + 2) * ts + base, nm * v.z);
    atomicAdd(AX + (size_t)(t0 + 3) * ts + base, nm * v.w);
  }
}

// ------------------------------ fused TGCN cell ----------------------------
// One wave per 16-node tile; 8 waves (256 threads) per block. N % 16 == 0.
__global__ void __launch_bounds__(256, 2)
tgcn_cell(const float* __restrict__ AXt, float* __restrict__ h,
          float* __restrict__ acc, const float* __restrict__ cwz,
          const float* __restrict__ cbz, const float* __restrict__ cwr,
          const float* __restrict__ cbr, const float* __restrict__ cwh,
          const float* __restrict__ cbh, const float* __restrict__ lwz,
          const float* __restrict__ lbz, const float* __restrict__ lwr,
          const float* __restrict__ lbr, const float* __restrict__ lwh,
          const float* __restrict__ lbh, const float* __restrict__ probs,
          int t, int n) {
  __shared__ float smem[8 * 16 * LDSS];
  const int lane = threadIdx.x & 31;
  const int col15 = lane & 15;
  const int hi = lane >> 4;
  const int wv = threadIdx.x >> 5;
  float* lds = smem + wv * 16 * LDSS;
  const int tile = blockIdx.x * 8 + wv;
  if (tile * 16 >= n) return;  // wave-uniform: EXEC stays all-ones for WMMA
  const int row0 = tile * 16;
  const int row = col15;
  const float* Atile = AXt + (size_t)row0 * F_IN;
  const float* hrow = h + (size_t)row0 * HID;

  // GCN gemm: [16x16] (AX tile) @ [16x32] W + bias -> D layout (c0: cols 0-15)
  auto gcn = [&](const float* W, const float* bias, v8f& c0, v8f& c1) {
    c0 = splat8(bias[col15]);
    c1 = splat8(bias[16 + col15]);
#pragma unroll
    for (int kb = 0; kb < 4; ++kb) {
      v2f a = a_chunk(Atile, F_IN, row, hi, kb);
      c0 = wmma4(a, b_chunk(W, kb, hi, col15), c0);
      c1 = wmma4(a, b_chunk(W, kb, hi, 16 + col15), c1);
    }
  };
  // D-layout -> LDS [16 x 32] tile (A-layout staging).
  auto to_lds = [&](const v8f& c0, const v8f& c1) {
    ds_fence();  // WAR vs. prior LDS reads (in-order DS, block IR reordering)
#pragma unroll
    for (int r = 0; r < 8; ++r) {
      const int m = r + 8 * hi;
      lds[m * LDSS + col15] = c0[r];
      lds[m * LDSS + 16 + col15] = c1[r];
    }
    ds_fence();  // RAW: reads below must see all lanes' stores
  };
  // Accumulate [16x32](LDS) @ [32x32] W into c0/c1.
  auto mm_lds = [&](const float* W, v8f& c0, v8f& c1) {
#pragma unroll
    for (int kb = 0; kb < 8; ++kb) {
      v2f a = a_chunk(lds, LDSS, row, hi, kb);
      c0 = wmma4(a, b_chunk(W, kb, hi, col15), c0);
      c1 = wmma4(a, b_chunk(W, kb, hi, 16 + col15), c1);
    }
  };
  // Accumulate [16x32](global, row-major) @ [32x32] W into c0/c1.
  auto mm_glb = [&](const float* A, const float* W, v8f& c0, v8f& c1) {
#pragma unroll
    for (int kb = 0; kb < 8; ++kb) {
      v2f a = a_chunk(A, HID, row, hi, kb);
      c0 = wmma4(a, b_chunk(W, kb, hi, col15), c0);
      c1 = wmma4(a, b_chunk(W, kb, hi, 16 + col15), c1);
    }
  };
  auto sig8 = [&](v8f& v) {
#pragma unroll
    for (int r = 0; r < 8; ++r) v[r] = 1.0f / (1.0f + __expf(-v[r]));
  };
  auto tanh8 = [&](v8f& v) {
#pragma unroll
    for (int r = 0; r < 8; ++r) v[r] = tanhf(v[r]);
  };

  v8f g0, g1;

  // ---- z gate: sigmoid([Gz, h] @ lwz + lbz) ----
  gcn(cwz, cbz, g0, g1);
  to_lds(g0, g1);
  v8f z0 = splat8(lbz[col15]), z1 = splat8(lbz[16 + col15]);
  mm_lds(lwz, z0, z1);
  mm_glb(hrow, lwz + 32 * HID, z0, z1);
  sig8(z0);
  sig8(z1);

  // ---- r gate ----
  gcn(cwr, cbr, g0, g1);
  to_lds(g0, g1);
  v8f r0 = splat8(lbr[col15]), r1 = splat8(lbr[16 + col15]);
  mm_lds(lwr, r0, r1);
  mm_glb(hrow, lwr + 32 * HID, r0, r1);
  sig8(r0);
  sig8(r1);

  // ---- h in D layout (for r*h and the GRU blend) ----
  v8f hd0, hd1;
#pragma unroll
  for (int r = 0; r < 8; ++r) {
    const int m = row0 + r + 8 * hi;
    hd0[r] = h[(size_t)m * HID + col15];
    hd1[r] = h[(size_t)m * HID + 16 + col15];
  }

  // ---- candidate: tanh([Gh, r*h] @ lwh + lbh) ----
  gcn(cwh, cbh, g0, g1);
  to_lds(g0, g1);
  v8f hc0 = splat8(lbh[col15]), hc1 = splat8(lbh[16 + col15]);
  mm_lds(lwh, hc0, hc1);
  v8f rh0, rh1;
#pragma unroll
  for (int r = 0; r < 8; ++r) {
    rh0[r] = r0[r] * hd0[r];
    rh1[r] = r1[r] * hd1[r];
  }
  to_lds(rh0, rh1);  // fences inside guard the Gh reads above
  mm_lds(lwh + 32 * HID, hc0, hc1);
  tanh8(hc0);
  tanh8(hc1);

  // ---- GRU blend, in-place h update, attention accumulation ----
  const float p = probs[t];
#pragma unroll
  for (int r = 0; r < 8; ++r) {
    const int m = row0 + r + 8 * hi;
    const size_t i0 = (size_t)m * HID + col15;
    const size_t i1 = i0 + 16;
    const float hn0 = z0[r] * hd0[r] + (1.0f - z0[r]) * hc0[r];
    const float hn1 = z1[r] * hd1[r] + (1.0f - z1[r]) * hc1[r];
    h[i0] = hn0;
    h[i1] = hn1;
    acc[i0] += p * hn0;
    acc[i1] += p * hn1;
  }
}

// ------------------------- head: relu -> 32x2 -> softmax -------------------
__global__ void head_softmax(const float* __restrict__ acc,
                             const float* __restrict__ hw,
                             const float* __restrict__ hb,
                             float* __restrict__ out, int n) {
  int i = blockIdx.x * blockDim.x + threadIdx.x;
  if (i >= n) return;
  const float* a = acc + (size_t)i * HID;
  float s0 = hb[0], s1 = hb[1];
#pragma unroll
  for (int f = 0; f < HID; ++f) {
    const float v = fmaxf(a[f], 0.0f);
    s0 += v * hw[f * 2 + 0];
    s1 += v * hw[f * 2 + 1];
  }
  const float m = fmaxf(s0, s1);
  const float e0 = __expf(s0 - m), e1 = __expf(s1 - m);
  const float inv = 1.0f / (e0 + e1);
  out[(size_t)i * 2 + 0] = e0 * inv;
  out[(size_t)i * 2 + 1] = e1 * inv;
}

// --------------------------------- launcher --------------------------------

extern "C" void kernel_launch(void* const* d_in, const int* in_sizes, int n_in,
                              void* d_out, int out_size, void* d_ws,
                              size_t ws_size, hipStream_t stream) {
  const float* x = (const float*)d_in[0];
  const int* ei = (const int*)d_in[1];
  const float* att = (const float*)d_in[2];
  const float* cwz = (const float*)d_in[3];
  const float* cbz = (const float*)d_in[4];
  const float* cwr = (const float*)d_in[5];
  const float* cbr = (const float*)d_in[6];
  const float* cwh = (const float*)d_in[7];
  const float* cbh = (const float*)d_in[8];
  const float* lwz = (const float*)d_in[9];
  const float* lbz = (const float*)d_in[10];
  const float* lwr = (const float*)d_in[11];
  const float* lbr = (const float*)d_in[12];
  const float* lwh = (const float*)d_in[13];
  const float* lbh = (const float*)d_in[14];
  const float* hw = (const float*)d_in[15];
  const float* hb = (const float*)d_in[16];
  float* out = (float*)d_out;

  const int T = in_sizes[2];
  const int N = in_sizes[0] / (F_IN * T);
  const long long E = (long long)in_sizes[1] / 2;
  const int* src = ei;
  const int* dst = ei + E;

  // Workspace layout (256B aligned slabs).
  char* wsp = (char*)d_ws;
  auto alloc = [&](size_t bytes) -> char* {
    char* p = wsp;
    wsp += (bytes + 255) & ~(size_t)255;
    return p;
  };
  float* deg = (float*)alloc((size_t)N * 4);
  float* dinv = (float*)alloc((size_t)N * 4);
  float* probs = (float*)alloc(256);
  float* AX = (float*)alloc((size_t)T * N * F_IN * 4);   // [T][N][16]
  float* h = (float*)alloc((size_t)N * HID * 4);
  float* acc = (float*)alloc((size_t)N * HID * 4);

  // Init: deg=1 (self loop), AX=0, h=0, acc=0.
  fill_f32<<<(N + 255) / 256, 256, 0, stream>>>(deg, 1.0f, N);
  const long long axn = (long long)T * N * F_IN;
  fill_f32<<<(int)((axn + 65535) / 65536), 256, 0, stream>>>(AX, 0.0f, axn);
  fill_f32<<<(N * HID + 65535) / 65536, 256, 0, stream>>>(h, 0.0f,
                                                          (long long)N * HID);
  fill_f32<<<(N * HID + 65535) / 65536, 256, 0, stream>>>(acc, 0.0f,
                                                          (long long)N * HID);
  attn_softmax_k<<<1, 32, 0, stream>>>(att, probs, T);

  // Degrees + D^-1/2.
  deg_accum<<<4096, 256, 0, stream>>>(dst, deg, E);
  rsqrt_k<<<(N + 255) / 256, 256, 0, stream>>>(deg, dinv, N);

  // One SpMM over all T*F_IN features.
  spmm_scatter<<<8192, 256, 0, stream>>>(src, dst, x, dinv, AX, E, N, T);

  // Sequential TGCN recurrence (12 fused WMMA cell launches).
  const int tiles = (N + 15) / 16;
  const int blocks = (tiles + 7) / 8;
  for (int t = 0; t < T; ++t) {
    tgcn_cell<<<blocks, 256, 0, stream>>>(AX + (size_t)t * N * F_IN, h, acc,
                                          cwz, cbz, cwr, cbr, cwh, cbh, lwz,
                                          lbz, lwr, lbr, lwh, lbh, probs, t, N);
  }

  // Head.
  head_softmax<<<(N + 255) / 256, 256, 0, stream>>>(acc, hw, hb, out, N);
}